// Capsule_24326694764663
// MI455X (gfx1250) — compile-verified
//
#include <hip/hip_runtime.h>
#include <math.h>

typedef __attribute__((ext_vector_type(2))) float v2f;
typedef __attribute__((ext_vector_type(8))) float v8f;

#define B_SZ     16
#define N_IN     1024
#define D_IN     256
#define NUM_CAPS 24
#define DIM_CAPS 128
#define N_OUT    (NUM_CAPS * DIM_CAPS)   /* 3072 */
#define M_ROWS   (B_SZ * N_IN)           /* 16384 */

// ---------------------------------------------------------------------------
// GEMM: U[16384,3072] = A[16384,256] * W[256,3072], exact fp32 via
// V_WMMA_F32_16X16X4_F32.  One wave computes a 16(M) x 64(N) strip:
// one A fragment reused across 4 B fragments per K-step of 4.
// A frag layout (ISA 7.12.2, 32-bit A 16x4): lane<16 -> {K0,K1} of row M=lane;
// lane>=16 -> {K2,K3} of row M=lane-16.  B frag mirrors with N = lane%16.
// C/D layout: VGPR i -> row (i + 8*(lane>=16)), col = lane%16.
// ---------------------------------------------------------------------------
__global__ __launch_bounds__(256) void caps_gemm(const float* __restrict__ A,
                                                 const float* __restrict__ W,
                                                 float* __restrict__ U) {
    const int lane   = threadIdx.x & 31;
    const int gw     = blockIdx.x * 8 + (threadIdx.x >> 5);
    const int mTile  = gw & 1023;   // 1024 tiles of 16 rows
    const int nStrip = gw >> 10;    // 48 strips of 64 cols
    const int half   = lane >> 4;
    const int r      = lane & 15;

    const float* arow = A + (size_t)(mTile * 16 + r) * D_IN + 2 * half;
    const float* bcol = W + (size_t)(2 * half) * N_OUT + nStrip * 64 + r;

    v8f acc0 = {}, acc1 = {}, acc2 = {}, acc3 = {};

#pragma unroll 4
    for (int k0 = 0; k0 < D_IN; k0 += 4) {
        v2f a;
        a.x = arow[k0];
        a.y = arow[k0 + 1];
        const float* bp = bcol + (size_t)k0 * N_OUT;
        v2f b0, b1, b2, b3;
        b0.x = bp[0];   b0.y = bp[N_OUT + 0];
        b1.x = bp[16];  b1.y = bp[N_OUT + 16];
        b2.x = bp[32];  b2.y = bp[N_OUT + 32];
        b3.x = bp[48];  b3.y = bp[N_OUT + 48];
        acc0 = __builtin_amdgcn_wmma_f32_16x16x4_f32(false, a, false, b0, (short)0, acc0, false, false);
        acc1 = __builtin_amdgcn_wmma_f32_16x16x4_f32(false, a, false, b1, (short)0, acc1, false, false);
        acc2 = __builtin_amdgcn_wmma_f32_16x16x4_f32(false, a, false, b2, (short)0, acc2, false, false);
        acc3 = __builtin_amdgcn_wmma_f32_16x16x4_f32(false, a, false, b3, (short)0, acc3, false, false);
    }

    float* crow = U + (size_t)(mTile * 16 + 8 * half) * N_OUT + nStrip * 64 + r;
#pragma unroll
    for (int i = 0; i < 8; ++i) {
        crow[(size_t)i * N_OUT +  0] = acc0[i];
        crow[(size_t)i * N_OUT + 16] = acc1[i];
        crow[(size_t)i * N_OUT + 32] = acc2[i];
        crow[(size_t)i * N_OUT + 48] = acc3[i];
    }
}

// ---------------------------------------------------------------------------
// s[b,k,d] = sum_n c[b,k,n] * u_hat[b,n,k,d]; then squash s/sqrt(||s||^2+eps).
// One block per (b,k), 128 threads (one per d).  uniform=1 -> c == 1/24.
// ---------------------------------------------------------------------------
__global__ __launch_bounds__(128) void caps_route_sum_squash(
    const float* __restrict__ U, const float* __restrict__ c, int uniform,
    float* __restrict__ out) {
    const int bk = blockIdx.x;          // 0..383
    const int d  = threadIdx.x;         // 0..127
    const int b  = bk / NUM_CAPS;
    const int k  = bk - b * NUM_CAPS;

    const float* base = U + ((size_t)b * N_IN) * N_OUT + (size_t)k * DIM_CAPS + d;
    float acc = 0.0f;
    if (uniform) {
#pragma unroll 8
        for (int n = 0; n < N_IN; ++n) acc += base[(size_t)n * N_OUT];
        acc *= (1.0f / (float)NUM_CAPS);
    } else {
        const float* cb = c + (size_t)bk * N_IN;
#pragma unroll 8
        for (int n = 0; n < N_IN; ++n)
            acc = fmaf(cb[n], base[(size_t)n * N_OUT], acc);
    }

    __shared__ float red[DIM_CAPS];
    red[d] = acc * acc;
    __syncthreads();
    for (int s = DIM_CAPS / 2; s > 0; s >>= 1) {
        if (d < s) red[d] += red[d + s];
        __syncthreads();
    }
    out[(size_t)bk * DIM_CAPS + d] = acc * (1.0f / sqrtf(red[0] + 1e-7f));
}

// ---------------------------------------------------------------------------
// logits[b,k,n] = dot_d(outputs[b,k,d], u_hat[b,n,k,d]).  Wave per (b,k,n),
// lane owns 4 d's (float4 coalesced), wave32 xor-shuffle reduction.
// ---------------------------------------------------------------------------
__global__ __launch_bounds__(256) void caps_agreement(
    const float* __restrict__ outv, const float* __restrict__ U,
    float* __restrict__ logits) {
    const int lane = threadIdx.x & 31;
    const int gw   = blockIdx.x * 8 + (threadIdx.x >> 5);  // 0..393215
    const int n    = gw & (N_IN - 1);
    const int bk   = gw >> 10;                             // 0..383
    const int b    = bk / NUM_CAPS;
    const int k    = bk - b * NUM_CAPS;

    const float4 u = *(const float4*)(U + ((size_t)b * N_IN + n) * N_OUT +
                                      (size_t)k * DIM_CAPS + lane * 4);
    const float4 o = *(const float4*)(outv + (size_t)bk * DIM_CAPS + lane * 4);
    float dot = u.x * o.x + u.y * o.y + u.z * o.z + u.w * o.w;
#pragma unroll
    for (int off = 16; off > 0; off >>= 1) dot += __shfl_xor(dot, off, 32);
    if (lane == 0) logits[(size_t)bk * N_IN + n] = dot;
}

// ---------------------------------------------------------------------------
// c[b,k,n] = softmax over k (24 capsules).  Thread per (b,n).
// ---------------------------------------------------------------------------
__global__ __launch_bounds__(256) void caps_softmax(const float* __restrict__ logits,
                                                    float* __restrict__ c) {
    const int t = blockIdx.x * 256 + threadIdx.x;  // 0..16383
    const int b = t >> 10;
    const int n = t & (N_IN - 1);
    const float* lb = logits + (size_t)b * NUM_CAPS * N_IN + n;

    float e[NUM_CAPS];
    float m = -3.402823466e38f;
#pragma unroll
    for (int k = 0; k < NUM_CAPS; ++k) {
        e[k] = lb[(size_t)k * N_IN];
        m = fmaxf(m, e[k]);
    }
    float s = 0.0f;
#pragma unroll
    for (int k = 0; k < NUM_CAPS; ++k) {
        e[k] = __expf(e[k] - m);
        s += e[k];
    }
    const float inv = 1.0f / s;
    float* cb = c + (size_t)b * NUM_CAPS * N_IN + n;
#pragma unroll
    for (int k = 0; k < NUM_CAPS; ++k) cb[(size_t)k * N_IN] = e[k] * inv;
}

// ---------------------------------------------------------------------------
extern "C" void kernel_launch(void* const* d_in, const int* in_sizes, int n_in,
                              void* d_out, int out_size, void* d_ws, size_t ws_size,
                              hipStream_t stream) {
    const float* u_vecs = (const float*)d_in[0];   // [16,1024,256]
    const float* W      = (const float*)d_in[1];   // [256,3072]
    float*       out    = (float*)d_out;           // [16,24,128]

    char* ws = (char*)d_ws;
    const size_t UHAT_B   = (size_t)M_ROWS * N_OUT * sizeof(float);      // 201326592
    const size_t LOGITS_B = (size_t)B_SZ * NUM_CAPS * N_IN * sizeof(float); // 1572864
    float* U      = (float*)ws;
    float* logits = (float*)(ws + UHAT_B);
    float* c      = (float*)(ws + UHAT_B + LOGITS_B);
    float* otmp   = (float*)(ws + UHAT_B + 2 * LOGITS_B);

    // u_hat = u_vecs @ W   (16384x256 * 256x3072): 49152 waves, 8 waves/block
    caps_gemm<<<6144, 256, 0, stream>>>(u_vecs, W, U);

    // routing iteration 0: b == 0 -> softmax is uniform 1/24
    caps_route_sum_squash<<<B_SZ * NUM_CAPS, DIM_CAPS, 0, stream>>>(U, (const float*)0, 1, otmp);

    // iterations 1 and 2
    for (int it = 0; it < 2; ++it) {
        caps_agreement<<<49152, 256, 0, stream>>>(otmp, U, logits);
        caps_softmax<<<64, 256, 0, stream>>>(logits, c);
        float* dst = (it == 1) ? out : otmp;
        caps_route_sum_squash<<<B_SZ * NUM_CAPS, DIM_CAPS, 0, stream>>>(U, c, 0, dst);
    }
}